// InfGenOccDecoder_26877905338814
// MI455X (gfx1250) — compile-verified
//
#include <hip/hip_runtime.h>
#include <math.h>

typedef __attribute__((ext_vector_type(2))) float v2f;
typedef __attribute__((ext_vector_type(8))) float v8f;

// Problem constants
#define NN    512
#define SS    18
#define VV    2048
#define HH    128
#define GRID_ 961
#define NH    961         // hidden width
#define K1    128         // feat inner dim
#define HSTR  977         // LDS row stride (961 padded to 976 + 1)
#define NHT   61          // ceil(961/16)
#define MROWS 2048        // feat rows
#define NOUT_A (3 * GRID_)   // 2883
#define NOUT_P (10 * GRID_)  // 9610
#define ASPLIT 2          // column groups for MLP A
#define PSPLIT 6          // column groups for MLP P

// ---------------------------------------------------------------------------
// Kernel 1: agent_pred_rel = token_traj @ rot(n,s) + prev_pos[:,:, -1,:]
// ---------------------------------------------------------------------------
__global__ __launch_bounds__(256)
void rotate_kernel(const float* __restrict__ traj,
                   const float* __restrict__ prev,
                   float* __restrict__ out) {
    int b = blockIdx.x;                 // n*S + s
    const float* pp = prev + (size_t)b * 4;   // [p0x,p0y,p1x,p1y]
    float dx = pp[2] - pp[0];
    float dy = pp[3] - pp[1];
    float h = atan2f(dy, dx);
    float s, c;
    __sincosf(h, &s, &c);
    float tx = pp[2], ty = pp[3];
    const float2* tin = (const float2*)(traj + (size_t)b * VV * 2);
    float2* tout = (float2*)(out + (size_t)b * VV * 2);
    for (int v = threadIdx.x; v < VV; v += 256) {
        float2 p = tin[v];
        float2 q;
        q.x =  p.x * c + p.y * s + tx;
        q.y = -p.x * s + p.y * c + ty;
        tout[v] = q;
    }
}

// ---------------------------------------------------------------------------
// Kernel 2: agent_token_emb gather.
// ---------------------------------------------------------------------------
__global__ __launch_bounds__(128)
void gather_kernel(const float* __restrict__ emb_veh,
                   const float* __restrict__ emb_ped,
                   const float* __restrict__ emb_cyc,
                   const int* __restrict__ tok_idx,
                   const int* __restrict__ atype,
                   float* __restrict__ out) {
    int b = blockIdx.x;                 // n*S + s
    int n = b / SS;
    int t = atype[n];
    const float* tab = (t == 0) ? emb_veh : (t == 1) ? emb_ped : emb_cyc;
    int idx = tok_idx[b];
    out[(size_t)b * HH + threadIdx.x] = tab[(size_t)idx * HH + threadIdx.x];
}

// ---------------------------------------------------------------------------
// Kernel 3: BOTH fused MLPs in one launch.
//   blockIdx.x : 16-row stripe of feat (128 stripes)
//   blockIdx.y : column group; y<ASPLIT -> MLP A, else MLP P
// Phase A (GEMM1+LN+ReLU) is ~1% of Phase B work, so recomputing it per
// column group is nearly free and buys 8x more blocks for machine fill.
//
// Branch-free out-of-range handling:
//  - B columns >= width: clamp the column pointer (col n of D only sees
//    col n of B; out-of-range D columns are never stored).
//  - K tail rows 961..963: LDS h is zero-padded there, A=0 annihilates
//    the finite clamped-B value.
// ---------------------------------------------------------------------------
__global__ __launch_bounds__(128)
__attribute__((amdgpu_waves_per_eu(4)))
void mlp_both_kernel(const float* __restrict__ feat,
                     const float* __restrict__ a_w1, const float* __restrict__ a_b1,
                     const float* __restrict__ a_g,  const float* __restrict__ a_b,
                     const float* __restrict__ a_w2, const float* __restrict__ a_b2,
                     const float* __restrict__ p_w1, const float* __restrict__ p_b1,
                     const float* __restrict__ p_g,  const float* __restrict__ p_b,
                     const float* __restrict__ p_w2, const float* __restrict__ p_b2,
                     float* __restrict__ out_a, float* __restrict__ out_p) {
    __shared__ float hlds[16 * HSTR];
    __shared__ float red[2 * 128];
    __shared__ float mu16[16];
    __shared__ float rs16[16];

    const int tid  = threadIdx.x;
    const int lane = tid & 31;
    const int wave = tid >> 5;
    const int m0   = blockIdx.x * 16;
    const int lm   = lane & 15;     // M / N index within tile
    const int hi   = lane >> 4;     // half-wave select
    const int kb   = hi * 2;        // K sub-offset for A/B fragments

    // ---- Select MLP (wave-uniform, from blockIdx.y) ----
    const bool is_p = (blockIdx.y >= ASPLIT);
    const int  grp    = is_p ? (int)blockIdx.y - ASPLIT : (int)blockIdx.y;
    const int  ngrps  = is_p ? PSPLIT : ASPLIT;
    const int  nout   = is_p ? NOUT_P : NOUT_A;
    const float* w1  = is_p ? p_w1 : a_w1;
    const float* b1  = is_p ? p_b1 : a_b1;
    const float* lng = is_p ? p_g  : a_g;
    const float* lnb = is_p ? p_b  : a_b;
    const float* w2  = is_p ? p_w2 : a_w2;
    const float* b2  = is_p ? p_b2 : a_b2;
    float* out       = is_p ? out_p : out_a;

    // ---------------- Phase A: h = feat @ w1 + b1 ----------------
    const float* arow = feat + (size_t)(m0 + lm) * K1;
    for (int t = wave; t < NHT; t += 4) {
        const int n0 = t * 16;
        const int nc = n0 + lm;
        const int ncc = (nc < NH) ? nc : (NH - 1);   // clamped column
        const float* w1c = w1 + ncc;
        v8f acc = {};
        #pragma unroll 4
        for (int k = 0; k < K1; k += 4) {
            v2f a, b;
            a.x = arow[k + kb];
            a.y = arow[k + kb + 1];
            b.x = w1c[(size_t)(k + kb) * NH];
            b.y = w1c[(size_t)(k + kb + 1) * NH];
            acc = __builtin_amdgcn_wmma_f32_16x16x4_f32(
                false, a, false, b, (short)0, acc, false, false);
        }
        const float bias = b1[ncc];
        #pragma unroll
        for (int r = 0; r < 8; ++r) {
            const int row = r + 8 * hi;
            hlds[row * HSTR + nc] = (nc < NH) ? (acc[r] + bias) : 0.0f;
        }
    }
    __syncthreads();

    // ---------------- LayerNorm stats: 8 threads per row ----------------
    {
        const int row = tid >> 3;
        const int sub = tid & 7;
        float s1 = 0.0f, s2 = 0.0f;
        for (int c = sub; c < NH; c += 8) {
            const float v = hlds[row * HSTR + c];
            s1 += v;
            s2 += v * v;
        }
        red[tid]       = s1;
        red[128 + tid] = s2;
        __syncthreads();
        if (sub == 0) {
            float a = 0.0f, b = 0.0f;
            #pragma unroll
            for (int i = 0; i < 8; ++i) {
                a += red[row * 8 + i];
                b += red[128 + row * 8 + i];
            }
            const float m   = a * (1.0f / (float)NH);
            const float var = b * (1.0f / (float)NH) - m * m;
            mu16[row] = m;
            rs16[row] = rsqrtf(var + 1e-5f);
        }
    }
    __syncthreads();

    // ---------------- LN apply + ReLU (pad cols stay 0) ----------------
    for (int i = tid; i < 16 * NH; i += 128) {
        const int row = i / NH;
        const int c   = i - row * NH;
        const float v = (hlds[row * HSTR + c] - mu16[row]) * rs16[row] * lng[c] + lnb[c];
        hlds[row * HSTR + c] = fmaxf(v, 0.0f);
    }
    __syncthreads();

    // ---------------- Phase B: out = h @ w2 + b2 (this col group) ----------
    const int NT2 = (nout + 15) >> 4;
    const int tpg = (NT2 + ngrps - 1) / ngrps;     // tiles per group
    const int t_begin = grp * tpg;
    const int t_end   = (t_begin + tpg < NT2) ? (t_begin + tpg) : NT2;
    const float* hrow = &hlds[lm * HSTR];
    for (int t = t_begin + wave; t < t_end; t += 4) {
        const int n0 = t * 16;
        const int nc = n0 + lm;
        const int ncc = (nc < nout) ? nc : (nout - 1);  // clamped column
        const float* w2c = w2 + ncc;
        v8f acc = {};
        // Main K loop: rows 0..959 (all in-range, unconditional)
        #pragma unroll 4
        for (int k = 0; k < 960; k += 4) {
            v2f a, b;
            a.x = hrow[k + kb];                 // single ds_load_b64
            a.y = hrow[k + kb + 1];
            b.x = w2c[(size_t)(k + kb) * nout];
            b.y = w2c[(size_t)(k + kb + 1) * nout];
            acc = __builtin_amdgcn_wmma_f32_16x16x4_f32(
                false, a, false, b, (short)0, acc, false, false);
        }
        // Tail: rows 960..963; A zero-padded past 960, B row clamped to 960
        {
            v2f a, b;
            a.x = hrow[960 + kb];
            a.y = hrow[960 + kb + 1];
            b.x = w2c[(size_t)960 * nout];
            b.y = b.x;
            acc = __builtin_amdgcn_wmma_f32_16x16x4_f32(
                false, a, false, b, (short)0, acc, false, false);
        }
        if (nc < nout) {
            const float bias = b2[nc];
            #pragma unroll
            for (int r = 0; r < 8; ++r) {
                const int row = m0 + r + 8 * hi;
                out[(size_t)row * nout + nc] = acc[r] + bias;
            }
        }
    }
}

// ---------------------------------------------------------------------------
// Host entry
// ---------------------------------------------------------------------------
extern "C" void kernel_launch(void* const* d_in, const int* in_sizes, int n_in,
                              void* d_out, int out_size, void* d_ws, size_t ws_size,
                              hipStream_t stream) {
    (void)in_sizes; (void)n_in; (void)out_size; (void)d_ws; (void)ws_size;

    const float* token_traj = (const float*)d_in[0];
    const float* prev_pos   = (const float*)d_in[1];
    const float* feat       = (const float*)d_in[2];
    const float* emb_veh    = (const float*)d_in[3];
    const float* emb_ped    = (const float*)d_in[4];
    const float* emb_cyc    = (const float*)d_in[5];
    const float* occ_a_w1   = (const float*)d_in[6];
    const float* occ_a_b1   = (const float*)d_in[7];
    const float* occ_a_g    = (const float*)d_in[8];
    const float* occ_a_b    = (const float*)d_in[9];
    const float* occ_a_w2   = (const float*)d_in[10];
    const float* occ_a_b2   = (const float*)d_in[11];
    const float* occ_p_w1   = (const float*)d_in[12];
    const float* occ_p_b1   = (const float*)d_in[13];
    const float* occ_p_g    = (const float*)d_in[14];
    const float* occ_p_b    = (const float*)d_in[15];
    const float* occ_p_w2   = (const float*)d_in[16];
    const float* occ_p_b2   = (const float*)d_in[17];
    const int*   tok_idx    = (const int*)d_in[18];
    const int*   atype      = (const int*)d_in[19];

    float* out = (float*)d_out;
    float* out_pred = out;                         // 512*18*2048*2 = 37,748,736
    float* out_emb  = out + 37748736;              // 512*18*128    =  1,179,648
    float* out_occa = out + 38928384;              // 2048*2883     =  5,904,384
    float* out_occp = out + 44832768;              // 2048*9610     = 19,681,280

    rotate_kernel<<<NN * SS, 256, 0, stream>>>(token_traj, prev_pos, out_pred);
    gather_kernel<<<NN * SS, 128, 0, stream>>>(emb_veh, emb_ped, emb_cyc,
                                               tok_idx, atype, out_emb);

    dim3 grid(MROWS / 16, ASPLIT + PSPLIT);        // 128 x 8 = 1024 blocks
    mlp_both_kernel<<<grid, 128, 0, stream>>>(
        feat,
        occ_a_w1, occ_a_b1, occ_a_g, occ_a_b, occ_a_w2, occ_a_b2,
        occ_p_w1, occ_p_b1, occ_p_g, occ_p_b, occ_p_w2, occ_p_b2,
        out_occa, out_occp);
}